// RoIAlignDeterministic_12996571038289
// MI455X (gfx1250) — compile-verified
//
#include <hip/hip_runtime.h>

typedef float    v2f __attribute__((ext_vector_type(2)));
typedef float    v4f __attribute__((ext_vector_type(4)));
typedef float    v8f __attribute__((ext_vector_type(8)));
typedef unsigned v4u __attribute__((ext_vector_type(4)));

#define FEAT_B 2
#define FEAT_C 256
#define FEAT_H 200
#define FEAT_W 336
#define OUTH 7
#define OUTW 7
#define NPIX 49
#define RSCALE 0.25f
#define CH_BYTES (FEAT_H * FEAT_W * 4)   // bytes per channel plane

// One block per ROI. 256 threads = 8 wave32.
// Phase 1: per-(pixel,sample) bilinear weights + byte gather-offsets in LDS;
//   x-clamp folded into the weights so B fragments are raw 8-byte pairs.
// Phase 2: scalar wave-tasks, OUT(16x16) = A(block-diag weights) x G(gathered)
//   via v_wmma_f32_16x16x4_f32. Gathers use GVS addressing (SGPR feat base +
//   32-bit voffset, vector-added to avoid per-load XCNT WAR waits), pipelined
//   2 groups (8 loads) ahead. Each lane loads its OWN pixel's 8 weights once
//   per task; per-group A is just a cndmask against the owning group id.
__global__ __launch_bounds__(256)
void roi_align_wmma_kernel(const float* __restrict__ feat,
                           const float* __restrict__ rois,
                           float* __restrict__ out,
                           int nrois) {
  __shared__ float    wq[64 * 16];       // [pixel][cy][sample][cx]  (4KB)
  __shared__ unsigned offq[2 * 64 * 4];  // [cy][pixel][sample] byte offsets (2KB)
  __shared__ int      s_bidx;

  const int n = blockIdx.x;
  if (n >= nrois) return;  // uniform
  const int tid = threadIdx.x;

  // ---------------- Phase 1: weights & offsets ----------------
  {
    const int p = tid >> 2;  // 0..63 (pixels 49..63 are zero padding)
    const int s = tid & 3;   // sub-sample (iy,ix)
    const float r0  = rois[n * 5 + 0];
    const float rx1 = rois[n * 5 + 1];
    const float ry1 = rois[n * 5 + 2];
    const float rx2 = rois[n * 5 + 3];
    const float ry2 = rois[n * 5 + 4];
    if (tid == 0) s_bidx = (int)r0;

    float wy0 = 0.f, wy1 = 0.f, wx0 = 0.f, wx1 = 0.f;
    unsigned o0 = 0u, o1 = 0u;
    if (p < NPIX) {
      const float sx = rx1 * RSCALE - 0.5f;
      const float sy = ry1 * RSCALE - 0.5f;
      const float ex = rx2 * RSCALE - 0.5f;
      const float ey = ry2 * RSCALE - 0.5f;
      const float bw = (ex - sx) / (float)OUTW;
      const float bh = (ey - sy) / (float)OUTH;
      const int ph = p / OUTW, pw = p % OUTW;
      const int iy = s >> 1, ix = s & 1;
      const float y = sy + ((float)ph + ((float)iy + 0.5f) * 0.5f) * bh;
      const float x = sx + ((float)pw + ((float)ix + 0.5f) * 0.5f) * bw;
      const bool valid = (y > -1.f) && (y < (float)FEAT_H) &&
                         (x > -1.f) && (x < (float)FEAT_W);
      float c0 = fmaxf(y, 0.f);
      int   lo = (int)floorf(c0);
      bool  hi = lo >= FEAT_H - 1;
      const int   ylo = hi ? FEAT_H - 1 : lo;
      const int   yhi = hi ? FEAT_H - 1 : lo + 1;
      const float ly  = hi ? 0.f : (c0 - (float)lo);
      c0 = fmaxf(x, 0.f);
      lo = (int)floorf(c0);
      hi = lo >= FEAT_W - 1;
      const int   xbase = hi ? FEAT_W - 2 : lo;  // pair = feat[y][xbase, xbase+1]
      const float lx    = hi ? 0.f : (c0 - (float)lo);

      const float vm = valid ? 0.25f : 0.f;  // 1/(SR*SR)
      wy0 = (1.f - ly) * vm;
      wy1 = ly * vm;
      wx0 = hi ? 0.f : (1.f - lx);  // clamped: weight moves to pair[1]
      wx1 = hi ? 1.f : lx;
      o0 = (unsigned)(ylo * FEAT_W + xbase) << 2;
      o1 = (unsigned)(yhi * FEAT_W + xbase) << 2;
    }
    const int qw = p * 16 + s * 2;   // [pixel][cy][sample][cx]
    wq[qw + 0] = wy0 * wx0;
    wq[qw + 1] = wy0 * wx1;
    wq[qw + 8] = wy1 * wx0;
    wq[qw + 9] = wy1 * wx1;
    offq[0 * 256 + p * 4 + s] = o0;
    offq[1 * 256 + p * 4 + s] = o1;
  }
  __syncthreads();

  // ---------------- Phase 2: WMMA tasks ----------------
  const int lane = tid & 31;
  const int wave = __builtin_amdgcn_readfirstlane(tid >> 5);   // scalar
  const int col  = lane & 15;        // N: channel-in-chunk; also A's row M
  const int cy   = (lane >> 4) & 1;  // 0: ylo row (K+0,1); 1: yhi row (K+2,3)
  const int bidx = __builtin_amdgcn_readfirstlane(s_bidx);     // scalar
  const unsigned colb = (unsigned)col * (unsigned)CH_BYTES;    // per-lane, hoisted
  const char* fbase = (const char*)feat;                       // uniform SGPR base

  // -------- 6 full-tile tasks (pt 0..2) --------
#pragma unroll 1
  for (int t = 0; t < 6; ++t) {
    const int ct    = ((t & 1) << 3) + wave;  // scalar channel chunk
    const int pt    = t >> 1;                 // scalar pixel tile
    const int pbase = pt * 16;
    const int c     = ct * 16 + col;
    const unsigned chanoff =
        colb + (unsigned)((bidx * FEAT_C + ct * 16)) * (unsigned)CH_BYTES;
    const unsigned* obase = offq + cy * 256 + pbase * 4;

    // this lane's OWN pixel weights (pixel pbase+col), loaded once per task
    const float* wown = wq + (pbase + col) * 16 + cy * 8;
    const v4f wA = *reinterpret_cast<const v4f*>(wown);
    const v4f wB = *reinterpret_cast<const v4f*>(wown + 4);

    auto gather = [&](int g, v2f* bs) {  // ds b128 offsets + 4 GVS b64 gathers
      const v4u o = *reinterpret_cast<const v4u*>(obase + g * 4);
      const v4u sp = {chanoff, chanoff, chanoff, chanoff};
      const v4u vo = o + sp;  // 4 distinct address regs -> no per-load XCNT WAR
#pragma unroll
      for (int s = 0; s < 4; ++s) {
        const float* fp = (const float*)(fbase + vo[s]);
        bs[s][0] = fp[0];  // merges into one global_load_b64
        bs[s][1] = fp[1];
      }
    };

    v2f b[3][4];  // rotating gather slots (registers)
    gather(0, b[0]);
    gather(1, b[1]);
    v8f acc = {};
#pragma unroll
    for (int g = 0; g < 16; ++g) {
      if (g + 2 < 16) gather(g + 2, b[(g + 2) % 3]);  // 8 loads in flight
      const bool own = (col == g);  // block-diagonal A
#pragma unroll
      for (int s = 0; s < 4; ++s) {
        const float w0 = (s < 2) ? wA[s * 2] : wB[(s - 2) * 2];
        const float w1 = (s < 2) ? wA[s * 2 + 1] : wB[(s - 2) * 2 + 1];
        v2f a;
        a[0] = own ? w0 : 0.f;
        a[1] = own ? w1 : 0.f;
        acc = __builtin_amdgcn_wmma_f32_16x16x4_f32(
            false, a, false, b[g % 3][s], (short)0, acc, false, false);
      }
    }

    // D: VGPR r -> pixel M = r + 8*cy, col N = channel; 8 contiguous floats
    float* outC = out + (size_t)(n * FEAT_C + c) * NPIX + pbase;
#pragma unroll
    for (int r = 0; r < 8; ++r) outC[r + cy * 8] = acc[r];
  }

  // -------- 2 tail tasks (pixel tile 3 = pixel 48 only, group 0) --------
  // Lanes col>0 read weights of padded pixels (all zero) -> no select needed.
#pragma unroll 1
  for (int t = 0; t < 2; ++t) {
    const int ct = (t << 3) + wave;
    const int c  = ct * 16 + col;
    const unsigned chanoff =
        colb + (unsigned)((bidx * FEAT_C + ct * 16)) * (unsigned)CH_BYTES;
    const unsigned* obase = offq + cy * 256 + 48 * 4;
    const float*    wown  = wq + (48 + col) * 16 + cy * 8;

    const v4u o  = *reinterpret_cast<const v4u*>(obase);
    const v4u sp = {chanoff, chanoff, chanoff, chanoff};
    const v4u vo = o + sp;
    const v4f wA = *reinterpret_cast<const v4f*>(wown);
    const v4f wB = *reinterpret_cast<const v4f*>(wown + 4);
    v8f acc = {};
#pragma unroll
    for (int s = 0; s < 4; ++s) {
      const float* fp = (const float*)(fbase + vo[s]);
      v2f b;
      b[0] = fp[0];
      b[1] = fp[1];
      v2f a;
      a[0] = (s < 2) ? wA[s * 2] : wB[(s - 2) * 2];
      a[1] = (s < 2) ? wA[s * 2 + 1] : wB[(s - 2) * 2 + 1];
      acc = __builtin_amdgcn_wmma_f32_16x16x4_f32(
          false, a, false, b, (short)0, acc, false, false);
    }
    if (lane < 16)
      out[(size_t)(n * FEAT_C + c) * NPIX + 48] = acc[0];
  }
}

extern "C" void kernel_launch(void* const* d_in, const int* in_sizes, int n_in,
                              void* d_out, int out_size, void* d_ws, size_t ws_size,
                              hipStream_t stream) {
  const float* feat = (const float*)d_in[0];
  const float* rois = (const float*)d_in[1];
  float* out = (float*)d_out;
  const int nrois = in_sizes[1] / 5;
  roi_align_wmma_kernel<<<nrois, 256, 0, stream>>>(feat, rois, out, nrois);
}